// SampleHistogramLoss_32444182954401
// MI455X (gfx1250) — compile-verified
//
#include <hip/hip_runtime.h>

typedef float v2f __attribute__((ext_vector_type(2)));
typedef float v8f __attribute__((ext_vector_type(8)));

#define NUM_BINS 128
#define DF 128
#define WS_FLOATS (2 * NUM_BINS + 2)   // [0..127] pos hist, [128..255] neg hist, [256] pos cnt, [257] neg cnt

// ---------------------------------------------------------------------------
// Kernel 0: zero the accumulation workspace (ws is poisoned, not re-zeroed by
// the harness between replays, so we must do it ourselves every call).
// ---------------------------------------------------------------------------
__global__ void hist_zero_ws(float* __restrict__ ws) {
    for (int i = threadIdx.x; i < WS_FLOATS; i += blockDim.x) ws[i] = 0.0f;
}

// ---------------------------------------------------------------------------
// Kernel 1: main streaming kernel.
// Each wave handles groups of 8 row-pairs. The 16x16 WMMA A-tile is packed as
// rows [a0..a7 ; b0..b7]; accumulating G = T x T^T over K=128 (32 WMMAs of
// 16x16x4 f32) yields a.a (lanes 0-7), a.b (lanes 8-23), b.b (lanes 24-31)
// at VGPR index (lane & 7) of the C/D layout.
// ---------------------------------------------------------------------------
__global__ __launch_bounds__(256)
void hist_main(const float* __restrict__ f0, const float* __restrict__ f1,
               const int* __restrict__ label, float* __restrict__ ws, int n) {
    __shared__ float sh[WS_FLOATS];
    for (int i = threadIdx.x; i < WS_FLOATS; i += blockDim.x) sh[i] = 0.0f;
    __syncthreads();

    const int lane          = threadIdx.x & 31;
    const int wave          = threadIdx.x >> 5;
    const int wavesPerBlock = blockDim.x >> 5;
    const int gwave         = blockIdx.x * wavesPerBlock + wave;
    const int nwaves        = gridDim.x * wavesPerBlock;
    const int ngroups       = n >> 3;          // 8 row-pairs per group

    // Packed-tile addressing: tile row m (0..15); m<8 -> feat_t0, m>=8 -> feat_t1.
    // A-operand 16x4 f32 layout: lanes 0-15 hold (K=0,K=1), lanes 16-31 hold (K=2,K=3).
    // The B-operand (4x16 = transpose of the same tile) has the identical per-lane
    // register layout, so one v2f per lane serves both operands.
    const int m    = lane & 15;
    const int koff = (lane >> 4) * 2;
    const float* const src = (m < 8) ? f0 : f1;
    const int selv = lane & 7;

    for (int g = gwave; g < ngroups; g += nwaves) {   // uniform per wave: EXEC stays full for WMMA
        const int base = g << 3;
        const float* p = src + (size_t)(base + (m & 7)) * DF + koff;

        v8f acc = {};
        #pragma unroll
        for (int k = 0; k < DF; k += 4) {
            v2f t = *reinterpret_cast<const v2f*>(p + k);   // global_load_b64, 8B aligned
            // G += T(16x4) x T^T(4x16)
            acc = __builtin_amdgcn_wmma_f32_16x16x4_f32(
                false, t, false, t, (short)0, acc, false, false);
        }

        // Pick this lane's Gram element: VGPR index = lane & 7 (unrolled cndmask chain).
        float gv = acc[0];
        if (selv == 1) gv = acc[1];
        if (selv == 2) gv = acc[2];
        if (selv == 3) gv = acc[3];
        if (selv == 4) gv = acc[4];
        if (selv == 5) gv = acc[5];
        if (selv == 6) gv = acc[6];
        if (selv == 7) gv = acc[7];

        // Gather the triple into lanes 0-7 (full-EXEC shuffles -> ds_bpermute).
        float ab = __shfl(gv,  8 + selv, 32);
        float bb = __shfl(gv, 24 + selv, 32);

        if (lane < 8) {
            const float aa  = gv;                     // lanes 0-7 hold a_i . a_i
            const int   row = base + lane;
            const float sim = ab * rsqrtf(aa * bb);   // cosine similarity
            float s = fminf(fmaxf(0.5f * (sim + 1.0f), 0.0f), 1.0f);
            float x = s * (float)(NUM_BINS - 1);      // s / STEP
            int   idx = (int)x;                       // floor (x >= 0)
            if (idx > NUM_BINS - 1) idx = NUM_BINS - 1;
            const float frac = x - (float)idx;

            const int lab = label[row];
            const int h   = (lab == 1) ? 0 : NUM_BINS;     // pos hist at [0], neg at [128]
            atomicAdd(&sh[h + idx], 1.0f - frac);          // ds_add_f32
            if (idx < NUM_BINS - 1) atomicAdd(&sh[h + idx + 1], frac);
            atomicAdd(&sh[2 * NUM_BINS + ((lab == 1) ? 0 : 1)], 1.0f);
        }
    }

    __syncthreads();
    for (int i = threadIdx.x; i < WS_FLOATS; i += blockDim.x)
        atomicAdd(&ws[i], sh[i]);                          // global_atomic_add_f32
}

// ---------------------------------------------------------------------------
// Kernel 2: normalize histograms, cumsum(pos) . neg -> scalar loss.
// ---------------------------------------------------------------------------
__global__ void hist_finalize(const float* __restrict__ ws, float* __restrict__ out) {
    __shared__ float hp[NUM_BINS];
    __shared__ float hn[NUM_BINS];
    const int t = threadIdx.x;
    const float pc = fmaxf(ws[2 * NUM_BINS],     1.0f);
    const float nc = fmaxf(ws[2 * NUM_BINS + 1], 1.0f);
    if (t < NUM_BINS) {
        hp[t] = ws[t] / pc;
        hn[t] = ws[NUM_BINS + t] / nc;
    }
    __syncthreads();
    if (t == 0) {
        float c = 0.0f, loss = 0.0f;
        for (int j = 0; j < NUM_BINS; ++j) { c += hp[j]; loss += c * hn[j]; }
        out[0] = loss;
    }
}

// ---------------------------------------------------------------------------
extern "C" void kernel_launch(void* const* d_in, const int* in_sizes, int n_in,
                              void* d_out, int out_size, void* d_ws, size_t ws_size,
                              hipStream_t stream) {
    const float* f0    = (const float*)d_in[0];
    const float* f1    = (const float*)d_in[1];
    const int*   label = (const int*)d_in[2];
    float*       ws    = (float*)d_ws;
    float*       out   = (float*)d_out;

    const int n = in_sizes[2];            // number of rows (labels are [N])
    const int ngroups = n >> 3;           // 8 row-pairs per wave-group
    const int wavesPerBlock = 8;          // 256 threads, wave32

    int blocks = (ngroups + wavesPerBlock - 1) / wavesPerBlock;
    if (blocks > 1024) blocks = 1024;     // grid-stride: ~4 groups/wave at N=262144
    if (blocks < 1)    blocks = 1;

    hist_zero_ws<<<1, 256, 0, stream>>>(ws);
    hist_main<<<blocks, 256, 0, stream>>>(f0, f1, label, ws, n);
    hist_finalize<<<1, 128, 0, stream>>>(ws, out);
}